// GCNLayer_34961033790071
// MI455X (gfx1250) — compile-verified
//
#include <hip/hip_runtime.h>

typedef __attribute__((ext_vector_type(2))) float v2f;
typedef __attribute__((ext_vector_type(8))) float v8f;

#define N_NODES 50000
#define N_EDGES 800000
#define CH      128
#define BN_EPS  1e-5f

// workspace layout (float offsets)
#define DEG_OFF 0        // 50000 floats: deg -> deg_inv_sqrt
#define SUM_OFF 50048    // 256 floats: per-channel sum, sumsq
#define XW_OFF  51200    // 6,400,000 floats: x @ W^T

// ---------------- init: deg = 1 (self loop), channel sums = 0 ----------------
__global__ void gcn_init_kernel(float* __restrict__ ws) {
    int i = blockIdx.x * blockDim.x + threadIdx.x;
    if (i < N_NODES) ws[DEG_OFF + i] = 1.0f;
    if (i < 2 * CH)  ws[SUM_OFF + i] = 0.0f;
}

// ---------------- degree count at targets ----------------
__global__ void gcn_degree_kernel(const long long* __restrict__ col,
                                  float* __restrict__ deg) {
    int e = blockIdx.x * blockDim.x + threadIdx.x;
    if (e < N_EDGES) {
        int c = (int)col[e];
        atomicAdd(&deg[c], 1.0f);
    }
}

__global__ void gcn_rsqrt_kernel(float* __restrict__ deg) {
    int i = blockIdx.x * blockDim.x + threadIdx.x;
    if (i < N_NODES) deg[i] = rsqrtf(deg[i]);   // deg >= 1 always
}

// ---------------- FP32 WMMA GEMM: xw = x @ W^T ----------------
// block = 256 threads = 8 waves; wave w handles M-tile (blockIdx.x*8 + w).
// B = W^T staged in LDS with K-pairs interleaved:
//   ldsB[(k>>1)*256 + n*2 + (k&1)] = B[k][n] = W[n*128 + k]
// so each B-fragment {B[k0][n], B[k0+1][n]} is ONE aligned ds_load_b64
// directly into the consecutive VGPR pair WMMA consumes (no repack movs).
__global__ __launch_bounds__(256) void gcn_gemm_wmma_kernel(
        const float* __restrict__ x, const float* __restrict__ W,
        float* __restrict__ xw) {
    __shared__ float ldsB[CH * CH];   // 64 KB

    for (int idx = threadIdx.x; idx < CH * CH; idx += 256) {
        int n = idx >> 7;          // W row  (out channel)
        int k = idx & 127;         // W col  (in channel)
        ldsB[(k >> 1) * 256 + n * 2 + (k & 1)] = W[idx];
    }
    __syncthreads();

    const int wave  = threadIdx.x >> 5;
    const int lane  = threadIdx.x & 31;
    const int tileM = blockIdx.x * 8 + wave;
    if (tileM >= (N_NODES / 16)) return;      // wave-uniform; EXEC stays all-1s

    const int half = lane >> 4;               // 0: lanes 0-15, 1: lanes 16-31
    const int mn   = lane & 15;               // M for A-frag, N for B-frag
    const int row  = tileM * 16 + mn;

    v8f acc[8] = {};                          // 16x16 f32 tiles for 8 N-tiles

    const float* xrow = x + row * CH;
    for (int kstep = 0; kstep < 32; ++kstep) {
        const int k0 = kstep * 4 + half * 2;          // K = k0, k0+1 here
        const v2f a = *(const v2f*)(xrow + k0);       // single b64 load

        const float* bbase = ldsB + (kstep * 2 + half) * 256 + mn * 2;
#pragma unroll
        for (int nt = 0; nt < 8; ++nt) {
            const v2f b = *(const v2f*)(bbase + nt * 32);   // single ds b64
            acc[nt] = __builtin_amdgcn_wmma_f32_16x16x4_f32(
                false, a, false, b, (short)0, acc[nt], false, false);
        }
    }

    // D layout: VGPR i -> M = i + 8*half, N = mn (per N-tile)
#pragma unroll
    for (int nt = 0; nt < 8; ++nt) {
#pragma unroll
        for (int i = 0; i < 8; ++i) {
            int r = tileM * 16 + half * 8 + i;
            int c = nt * 16 + mn;
            xw[r * CH + c] = acc[nt][i];
        }
    }
}

// ---------------- out = dinv^2 * xw  + bias  (self-loop term) ----------------
__global__ void gcn_selfloop_bias_kernel(const float* __restrict__ xw,
                                         const float* __restrict__ dinv,
                                         const float* __restrict__ bias,
                                         float* __restrict__ out) {
    int i = blockIdx.x * blockDim.x + threadIdx.x;
    if (i < N_NODES * CH) {
        int node = i >> 7;
        int ch   = i & 127;
        float d  = dinv[node];
        out[i] = d * d * xw[i] + bias[ch];
    }
}

// ---------------- edge scatter-add (L2-resident f32 atomics) ----------------
__global__ void gcn_scatter_kernel(const long long* __restrict__ rowp,
                                   const long long* __restrict__ colp,
                                   const float* __restrict__ xw,
                                   const float* __restrict__ dinv,
                                   float* __restrict__ out) {
    int idx = blockIdx.x * blockDim.x + threadIdx.x;
    if (idx < N_EDGES * CH) {
        int e  = idx >> 7;
        int ch = idx & 127;
        int r  = (int)rowp[e];
        int c  = (int)colp[e];
        float norm = dinv[r] * dinv[c];
        atomicAdd(&out[c * CH + ch], norm * xw[r * CH + ch]);
    }
}

// ---------------- per-channel sum / sumsq (block-local then atomic) ----------
__global__ __launch_bounds__(128) void gcn_stats_kernel(
        const float* __restrict__ out, float* __restrict__ sums) {
    int ch = threadIdx.x;                 // 128 threads = 128 channels
    int r0 = blockIdx.x * 256;
    int r1 = r0 + 256; if (r1 > N_NODES) r1 = N_NODES;
    float s = 0.0f, sq = 0.0f;
    for (int r = r0; r < r1; ++r) {
        float v = out[r * CH + ch];       // coalesced across channels
        s += v;
        sq += v * v;
    }
    atomicAdd(&sums[ch], s);
    atomicAdd(&sums[CH + ch], sq);
}

// ---------------- BN (batch stats) + affine + ReLU, in place ----------------
__global__ void gcn_bn_relu_kernel(float* __restrict__ out,
                                   const float* __restrict__ sums,
                                   const float* __restrict__ gamma,
                                   const float* __restrict__ beta) {
    int i = blockIdx.x * blockDim.x + threadIdx.x;
    if (i < N_NODES * CH) {
        int ch = i & 127;
        const float invN = 1.0f / (float)N_NODES;
        float mean = sums[ch] * invN;
        float var  = sums[CH + ch] * invN - mean * mean;
        float inv  = rsqrtf(var + BN_EPS);
        float v    = (out[i] - mean) * inv * gamma[ch] + beta[ch];
        out[i] = fmaxf(v, 0.0f);
    }
}

extern "C" void kernel_launch(void* const* d_in, const int* in_sizes, int n_in,
                              void* d_out, int out_size, void* d_ws, size_t ws_size,
                              hipStream_t stream) {
    const float*     x     = (const float*)d_in[0];
    const long long* ei    = (const long long*)d_in[1];   // int64 [2, E]
    const float*     W     = (const float*)d_in[2];
    const float*     bias  = (const float*)d_in[3];
    const float*     gamma = (const float*)d_in[4];
    const float*     beta  = (const float*)d_in[5];
    float* out = (float*)d_out;
    float* ws  = (float*)d_ws;

    float* deg  = ws + DEG_OFF;   // becomes deg_inv_sqrt
    float* sums = ws + SUM_OFF;
    float* xw   = ws + XW_OFF;

    const long long* rowp = ei;             // sources
    const long long* colp = ei + N_EDGES;   // targets

    gcn_init_kernel<<<(N_NODES + 255) / 256, 256, 0, stream>>>(ws);
    gcn_degree_kernel<<<(N_EDGES + 255) / 256, 256, 0, stream>>>(colp, deg);
    gcn_rsqrt_kernel<<<(N_NODES + 255) / 256, 256, 0, stream>>>(deg);

    // 3125 M-tiles of 16 rows, 8 tiles per block -> 391 blocks
    gcn_gemm_wmma_kernel<<<(N_NODES / 16 + 7) / 8, 256, 0, stream>>>(x, W, xw);

    gcn_selfloop_bias_kernel<<<(N_NODES * CH + 255) / 256, 256, 0, stream>>>(
        xw, deg, bias, out);
    gcn_scatter_kernel<<<(N_EDGES * CH + 255) / 256, 256, 0, stream>>>(
        rowp, colp, xw, deg, out);

    gcn_stats_kernel<<<(N_NODES + 255) / 256, 128, 0, stream>>>(out, sums);
    gcn_bn_relu_kernel<<<(N_NODES * CH + 255) / 256, 256, 0, stream>>>(
        out, sums, gamma, beta);
}